// GraphAttentionLayer_75900662055042
// MI455X (gfx1250) — compile-verified
//
#include <hip/hip_runtime.h>
#include <math.h>

// ---- problem constants (match reference) ----
#define N_NODES 50000
#define N_EDGES 800000
#define IN_F    256
#define HEADS   4
#define CDIM    32
#define OUT_F   128          // HEADS * CDIM
#define NEG_SLOPE 0.2f
#define BN_EPS  1e-5f
#define E_TOT   (N_EDGES + N_NODES)   // edges + self loops

typedef float v2f __attribute__((ext_vector_type(2)));
typedef float v8f __attribute__((ext_vector_type(8)));

#define LDS_STRIDE 257   // 256 + 1 pad -> conflict-free column reads across 64 banks

// ---------------------------------------------------------------------------
// Kernel 1: h = x @ W  via V_WMMA_F32_16X16X4_F32 (fp32 WMMA, wave32)
// Block: 256 threads = 8 waves. Block computes a 16-row x 128-col strip.
// Wave w computes the 16x16 tile at columns [16w, 16w+16).
// ---------------------------------------------------------------------------
__global__ __launch_bounds__(256) void gat_gemm_wmma(
    const float* __restrict__ x, const float* __restrict__ W,
    float* __restrict__ h)
{
    __shared__ float xs[16 * LDS_STRIDE];

    const int tid  = threadIdx.x;
    const int lane = tid & 31;
    const int wave = tid >> 5;            // 0..7 -> output column tile
    const int row0 = blockIdx.x * 16;

    // Cooperative load of the A tile: 16 rows x 256 cols of x into LDS.
    for (int i = tid; i < 16 * IN_F; i += 256) {
        const int r = i >> 8;             // / 256
        const int k = i & 255;
        float v = 0.0f;
        const int gr = row0 + r;
        if (gr < N_NODES) v = x[(size_t)gr * IN_F + k];
        xs[r * LDS_STRIDE + k] = v;
    }
    __syncthreads();

    // Fragment indices per ISA layout:
    //  A 16x4 f32: lanes 0-15 -> M=lane, K={k0, k0+1}; lanes 16-31 -> M=lane-16, K={k0+2, k0+3}
    //  B 4x16 f32 mirrors the K split, N = lane&15
    const int m    = lane & 15;
    const int koff = (lane >> 4) * 2;
    const int col  = wave * 16 + m;       // output column this lane owns in C/D

    v8f acc = {0.f,0.f,0.f,0.f,0.f,0.f,0.f,0.f};

    #pragma unroll
    for (int k0 = 0; k0 < IN_F; k0 += 4) {
        v2f a, b;
        a.x = xs[m * LDS_STRIDE + k0 + koff];
        a.y = xs[m * LDS_STRIDE + k0 + koff + 1];
        b.x = W[(size_t)(k0 + koff)     * OUT_F + col];
        b.y = W[(size_t)(k0 + koff + 1) * OUT_F + col];
        acc = __builtin_amdgcn_wmma_f32_16x16x4_f32(
            /*neg_a=*/false, a, /*neg_b=*/false, b,
            /*c_mod=*/(short)0, acc, /*reuse_a=*/false, /*reuse_b=*/false);
    }

    // C/D layout: VGPR i -> M = i (lanes 0-15) or i+8 (lanes 16-31), N = lane&15
    const int rbase = row0 + ((lane >> 4) << 3);
    #pragma unroll
    for (int i = 0; i < 8; ++i) {
        const int r = rbase + i;
        if (r < N_NODES) h[(size_t)r * OUT_F + col] = acc[i];
    }
}

// ---------------------------------------------------------------------------
// Kernel 2: per-(node, head) attention coefficients a_src, a_dst
// ---------------------------------------------------------------------------
__global__ void gat_attn_coef(const float* __restrict__ h,
                              const float* __restrict__ att_src,
                              const float* __restrict__ att_dst,
                              float* __restrict__ a_src,
                              float* __restrict__ a_dst)
{
    const int idx = blockIdx.x * blockDim.x + threadIdx.x;   // node*HEADS + head
    if (idx >= N_NODES * HEADS) return;
    const int n  = idx >> 2;
    const int hd = idx & 3;
    const float* hp = h + (size_t)n * OUT_F + hd * CDIM;
    const float* as = att_src + hd * CDIM;
    const float* ad = att_dst + hd * CDIM;
    float s = 0.f, d = 0.f;
    #pragma unroll
    for (int c = 0; c < CDIM; ++c) {
        const float v = hp[c];
        s += v * as[c];
        d += v * ad[c];
    }
    a_src[idx] = s;
    a_dst[idx] = d;
}

// ---------------------------------------------------------------------------
// Kernel 3: init accumulators (out=0, emax=-inf, denom=0, bn sums=0)
// ---------------------------------------------------------------------------
__global__ void gat_init(float* __restrict__ out, float* __restrict__ emax,
                         float* __restrict__ denom, float* __restrict__ bnacc)
{
    const int i = blockIdx.x * blockDim.x + threadIdx.x;
    if (i < N_NODES * OUT_F) out[i] = 0.0f;
    if (i < N_NODES * HEADS) { emax[i] = -INFINITY; denom[i] = 0.0f; }
    if (i < 2 * OUT_F) bnacc[i] = 0.0f;
}

__device__ __forceinline__ void edge_pair(const long long* __restrict__ ei,
                                          int e, int& s, int& d)
{
    if (e < N_EDGES) { s = (int)ei[e]; d = (int)ei[N_EDGES + e]; }
    else             { s = d = e - N_EDGES; }       // self loops
}

// ---------------------------------------------------------------------------
// Kernel 4: segment max of leaky-relu'd logits over destinations
// ---------------------------------------------------------------------------
__global__ void gat_edge_max(const long long* __restrict__ ei,
                             const float* __restrict__ a_src,
                             const float* __restrict__ a_dst,
                             float* __restrict__ emax)
{
    const int idx = blockIdx.x * blockDim.x + threadIdx.x;  // edge*HEADS + head
    if (idx >= E_TOT * HEADS) return;
    const int e  = idx >> 2;
    const int hd = idx & 3;
    int s, d; edge_pair(ei, e, s, d);
    float v = a_src[s * HEADS + hd] + a_dst[d * HEADS + hd];
    v = (v > 0.f) ? v : v * NEG_SLOPE;
    atomicMax(&emax[d * HEADS + hd], v);   // GLOBAL_ATOMIC_MAX_NUM_F32
}

// ---------------------------------------------------------------------------
// Kernel 5: segment sum of exp(e - emax) over destinations
// ---------------------------------------------------------------------------
__global__ void gat_edge_sum(const long long* __restrict__ ei,
                             const float* __restrict__ a_src,
                             const float* __restrict__ a_dst,
                             const float* __restrict__ emax,
                             float* __restrict__ denom)
{
    const int idx = blockIdx.x * blockDim.x + threadIdx.x;
    if (idx >= E_TOT * HEADS) return;
    const int e  = idx >> 2;
    const int hd = idx & 3;
    int s, d; edge_pair(ei, e, s, d);
    float v = a_src[s * HEADS + hd] + a_dst[d * HEADS + hd];
    v = (v > 0.f) ? v : v * NEG_SLOPE;
    const float ex = __expf(v - emax[d * HEADS + hd]);
    atomicAdd(&denom[d * HEADS + hd], ex);
}

// ---------------------------------------------------------------------------
// Kernel 6: weighted scatter: out[dst] += alpha * h[src]. One wave per edge;
// lane owns 4 consecutive features (one head per 8 lanes).
// ---------------------------------------------------------------------------
__global__ __launch_bounds__(256) void gat_scatter(
    const long long* __restrict__ ei, const float* __restrict__ h,
    const float* __restrict__ a_src, const float* __restrict__ a_dst,
    const float* __restrict__ emax, const float* __restrict__ denom,
    float* __restrict__ out)
{
    const int gid  = blockIdx.x * blockDim.x + threadIdx.x;
    const int e    = gid >> 5;
    const int lane = gid & 31;
    if (e >= E_TOT) return;
    int s, d; edge_pair(ei, e, s, d);

    const int f0 = lane * 4;         // features [f0, f0+4)
    const int hd = f0 >> 5;          // head for this lane

    float v = a_src[s * HEADS + hd] + a_dst[d * HEADS + hd];
    v = (v > 0.f) ? v : v * NEG_SLOPE;
    const float alpha = __expf(v - emax[d * HEADS + hd])
                        / (denom[d * HEADS + hd] + 1e-16f);

    const float4 hv = *reinterpret_cast<const float4*>(h + (size_t)s * OUT_F + f0);
    float* op = out + (size_t)d * OUT_F + f0;
    atomicAdd(op + 0, hv.x * alpha);
    atomicAdd(op + 1, hv.y * alpha);
    atomicAdd(op + 2, hv.z * alpha);
    atomicAdd(op + 3, hv.w * alpha);
}

// ---------------------------------------------------------------------------
// Kernel 7: BN statistics (per-feature sum and sum of squares over nodes)
// ---------------------------------------------------------------------------
#define BN_NODES_PER_BLOCK 128
__global__ __launch_bounds__(128) void gat_bn_stats(
    const float* __restrict__ out, const float* __restrict__ bias,
    float* __restrict__ bnacc)        // bnacc[0:128]=sum, bnacc[128:256]=sumsq
{
    const int f  = threadIdx.x;
    const int n0 = blockIdx.x * BN_NODES_PER_BLOCK;
    const float b = bias[f];
    float s = 0.f, sq = 0.f;
    for (int i = 0; i < BN_NODES_PER_BLOCK; ++i) {
        const int n = n0 + i;
        if (n < N_NODES) {
            const float v = out[(size_t)n * OUT_F + f] + b;
            s  += v;
            sq += v * v;
        }
    }
    atomicAdd(&bnacc[f], s);
    atomicAdd(&bnacc[OUT_F + f], sq);
}

// ---------------------------------------------------------------------------
// Kernel 8: apply bias + BN (batch stats, biased var) + ELU, in place
// ---------------------------------------------------------------------------
__global__ void gat_bn_elu(float* __restrict__ out,
                           const float* __restrict__ bias,
                           const float* __restrict__ gamma,
                           const float* __restrict__ beta,
                           const float* __restrict__ bnacc)
{
    const int i = blockIdx.x * blockDim.x + threadIdx.x;
    if (i >= N_NODES * OUT_F) return;
    const int f = i & (OUT_F - 1);
    const float invN = 1.0f / (float)N_NODES;
    const float mean = bnacc[f] * invN;
    const float var  = bnacc[OUT_F + f] * invN - mean * mean;
    const float v    = out[i] + bias[f];
    float y = (v - mean) * rsqrtf(var + BN_EPS) * gamma[f] + beta[f];
    y = (y > 0.f) ? y : (__expf(y) - 1.0f);   // ELU(alpha=1)
    out[i] = y;
}

// ---------------------------------------------------------------------------
// Host launcher
// ---------------------------------------------------------------------------
extern "C" void kernel_launch(void* const* d_in, const int* in_sizes, int n_in,
                              void* d_out, int out_size, void* d_ws, size_t ws_size,
                              hipStream_t stream)
{
    const float*     x        = (const float*)d_in[0];
    const long long* ei       = (const long long*)d_in[1];   // int64 edge_index [2, E]
    const float*     W        = (const float*)d_in[2];
    const float*     att_src  = (const float*)d_in[3];
    const float*     att_dst  = (const float*)d_in[4];
    const float*     bias     = (const float*)d_in[5];
    const float*     gamma    = (const float*)d_in[6];
    const float*     beta     = (const float*)d_in[7];
    float*           out      = (float*)d_out;

    // Workspace layout (floats), 256B-aligned sections
    char* ws = (char*)d_ws;
    size_t off = 0;
    auto take = [&](size_t nfloats) {
        char* p = ws + off;
        off += ((nfloats * sizeof(float)) + 255) & ~(size_t)255;
        return (float*)p;
    };
    float* h      = take((size_t)N_NODES * OUT_F);    // 25.6 MB
    float* a_src  = take((size_t)N_NODES * HEADS);
    float* a_dst  = take((size_t)N_NODES * HEADS);
    float* emax   = take((size_t)N_NODES * HEADS);
    float* denom  = take((size_t)N_NODES * HEADS);
    float* bnacc  = take(2 * OUT_F);
    (void)ws_size; (void)in_sizes; (void)n_in; (void)out_size;

    // 1) GEMM via WMMA: 3125 row tiles of 16, 8 waves cover 128 cols
    gat_gemm_wmma<<<(N_NODES + 15) / 16, 256, 0, stream>>>(x, W, h);

    // 2) attention coefficients
    {
        const int n = N_NODES * HEADS;
        gat_attn_coef<<<(n + 255) / 256, 256, 0, stream>>>(h, att_src, att_dst, a_src, a_dst);
    }

    // 3) init accumulators
    {
        const int n = N_NODES * OUT_F;
        gat_init<<<(n + 255) / 256, 256, 0, stream>>>(out, emax, denom, bnacc);
    }

    // 4) segment max
    {
        const int n = E_TOT * HEADS;
        gat_edge_max<<<(n + 255) / 256, 256, 0, stream>>>(ei, a_src, a_dst, emax);
    }

    // 5) segment sum of exp
    {
        const int n = E_TOT * HEADS;
        gat_edge_sum<<<(n + 255) / 256, 256, 0, stream>>>(ei, a_src, a_dst, emax, denom);
    }

    // 6) weighted scatter (one wave per edge)
    {
        const long long threads = (long long)E_TOT * 32;
        const int blocks = (int)((threads + 255) / 256);
        gat_scatter<<<blocks, 256, 0, stream>>>(ei, h, a_src, a_dst, emax, denom, out);
    }

    // 7) BN stats
    {
        const int blocks = (N_NODES + BN_NODES_PER_BLOCK - 1) / BN_NODES_PER_BLOCK;
        gat_bn_stats<<<blocks, 128, 0, stream>>>(out, bias, bnacc);
    }

    // 8) BN apply + ELU
    {
        const int n = N_NODES * OUT_F;
        gat_bn_elu<<<(n + 255) / 256, 256, 0, stream>>>(out, bias, gamma, beta, bnacc);
    }
}